// MedianPool2d_18554258719063
// MI455X (gfx1250) — compile-verified
//
#include <hip/hip_runtime.h>
#include <cstdint>
#include <cstddef>

// MedianPool2d 3x3, stride 1, pad 1 on float32[8,32,512,512].
// Memory-bound: 537 MB traffic -> ~23us floor at 23.3 TB/s.
// CDNA5 path: global_load_async_to_lds_b128 row staging (ASYNCcnt),
// 16-slot LDS ring, 4-row groups double-buffered (8 rows / 16KB in flight
// per block to cover HBM bandwidth-delay product), med3/min3/max3 network.

constexpr int Hh     = 512;
constexpr int Ww     = 512;
constexpr int PLANES = 8 * 32;   // B*C
constexpr int SH     = 64;       // output rows per block strip
constexpr int G      = 4;        // rows per pipeline stage
constexpr int RB     = 16;       // ring slots (power of two)
constexpr int TPB    = 128;      // threads per block = Ww/4

typedef float v4f __attribute__((ext_vector_type(4)));
typedef int   v4i __attribute__((ext_vector_type(4)));
typedef __attribute__((address_space(1))) v4i* gv4i_ptr;   // global int4*
typedef __attribute__((address_space(3))) v4i* lv4i_ptr;   // LDS int4*

#if defined(__HIP_DEVICE_COMPILE__) && __has_builtin(__builtin_amdgcn_global_load_async_to_lds_b128)
#define USE_ASYNC 1
#else
#define USE_ASYNC 0
#endif

__device__ __forceinline__ float f_min3(float a, float b, float c) {
    return fminf(fminf(a, b), c);
}
__device__ __forceinline__ float f_max3(float a, float b, float c) {
    return fmaxf(fmaxf(a, b), c);
}
__device__ __forceinline__ float f_med3(float a, float b, float c) {
#if defined(__HIP_DEVICE_COMPILE__) && __has_builtin(__builtin_amdgcn_fmed3f)
    return __builtin_amdgcn_fmed3f(a, b, c);
#else
    return fmaxf(fminf(fmaxf(a, b), c), fminf(a, b));
#endif
}

// Wait until at most N async-LDS loads remain outstanding (per wave).
__device__ __forceinline__ void wait_async_le4() {
#if USE_ASYNC
#if __has_builtin(__builtin_amdgcn_s_wait_asynccnt)
    __builtin_amdgcn_s_wait_asynccnt(4);
#else
    asm volatile("s_wait_asynccnt 0x4" ::: "memory");
#endif
#endif
}

// Stage one 512-float input row into ring slot (row & 15). The row index is
// clamped to [0, rmax] so every stage is exactly one async instruction per
// wave (uniform ASYNCcnt bookkeeping); pad rows are substituted with zeros at
// compute time instead. Each of 128 threads moves 16B -> 2KB per row.
__device__ __forceinline__ void stage_row(const float* __restrict__ xp,
                                          float (*ring)[Ww], int row, int rmax,
                                          int t) {
    int rc = row < 0 ? 0 : (row > rmax ? rmax : row);
    float*       dst = ring[row & (RB - 1)] + 4 * t;
    const float* g   = xp + (size_t)rc * Ww + 4 * t;
#if USE_ASYNC
    __builtin_amdgcn_global_load_async_to_lds_b128(
        (gv4i_ptr)g, (lv4i_ptr)dst, /*offset=*/0, /*cpol=*/0);
#else
    *(v4f*)dst = *(const v4f*)g;
#endif
}

__global__ void __launch_bounds__(TPB)
median3x3_kernel(const float* __restrict__ x, float* __restrict__ out) {
    const int plane = blockIdx.x;            // 0..255  (B*C planes)
    const int r0    = blockIdx.y * SH;       // first output row of strip
    const int r1    = r0 + SH;               // one past last output row
    const int rmax  = (r1 < Hh ? r1 : Hh - 1);
    const int t     = threadIdx.x;           // 0..127

    const float* xp = x   + (size_t)plane * Hh * Ww;
    float*       op = out + (size_t)plane * Hh * Ww;

    __shared__ float ring[RB][Ww];           // 32 KB ring of input rows

    // Prologue: rows r0-1 .. r0+2G (group-0 needs 6 rows, plus group-1's 4).
#pragma unroll
    for (int k = -1; k <= 2 * G; ++k)
        stage_row(xp, ring, r0 + k, rmax, t);
    wait_async_le4();                        // rows r0-1..r0+G landed
    __syncthreads();

    const int c0 = 4 * t;                    // base output column
    for (int gbase = r0; gbase < r1; gbase += G) {
        // Prefetch the group-after-next: rows gbase+2G+1 .. gbase+3G.
#pragma unroll
        for (int k = 1; k <= G; ++k)
            stage_row(xp, ring, gbase + 2 * G + k, rmax, t);

        // Compute output rows gbase .. gbase+G-1 from landed ring slots.
#pragma unroll
        for (int i = 0; i < G; ++i) {
            const int r = gbase + i;
            const float* Lm = ring[(r - 1) & (RB - 1)];
            const float* L0 = ring[(r    ) & (RB - 1)];
            const float* Lp = ring[(r + 1) & (RB - 1)];

            float A[6], Bv[6], Cv[6];

            v4f vb = *(const v4f*)(L0 + c0);
            Bv[0] = (t == 0)       ? 0.f : L0[c0 - 1];
            Bv[5] = (t == TPB - 1) ? 0.f : L0[c0 + 4];
            Bv[1] = vb.x; Bv[2] = vb.y; Bv[3] = vb.z; Bv[4] = vb.w;

            if (r > 0) {                     // top neighbor row (zero-pad at r==0)
                v4f va = *(const v4f*)(Lm + c0);
                A[0] = (t == 0)       ? 0.f : Lm[c0 - 1];
                A[5] = (t == TPB - 1) ? 0.f : Lm[c0 + 4];
                A[1] = va.x; A[2] = va.y; A[3] = va.z; A[4] = va.w;
            } else {
#pragma unroll
                for (int j = 0; j < 6; ++j) A[j] = 0.f;
            }

            if (r < Hh - 1) {                // bottom neighbor row (zero-pad at r==H-1)
                v4f vc = *(const v4f*)(Lp + c0);
                Cv[0] = (t == 0)       ? 0.f : Lp[c0 - 1];
                Cv[5] = (t == TPB - 1) ? 0.f : Lp[c0 + 4];
                Cv[1] = vc.x; Cv[2] = vc.y; Cv[3] = vc.z; Cv[4] = vc.w;
            } else {
#pragma unroll
                for (int j = 0; j < 6; ++j) Cv[j] = 0.f;
            }

            // Vertical sorted triple per column (shared by 3 horizontal windows).
            float mn[6], md[6], mx[6];
#pragma unroll
            for (int j = 0; j < 6; ++j) {
                mn[j] = f_min3(A[j], Bv[j], Cv[j]);
                md[j] = f_med3(A[j], Bv[j], Cv[j]);
                mx[j] = f_max3(A[j], Bv[j], Cv[j]);
            }

            // median9 = med3( max3(col mins), med3(col meds), min3(col maxes) )
            v4f o;
#pragma unroll
            for (int k = 0; k < 4; ++k) {
                float lo = f_max3(mn[k], mn[k + 1], mn[k + 2]);
                float mi = f_med3(md[k], md[k + 1], md[k + 2]);
                float hi = f_min3(mx[k], mx[k + 1], mx[k + 2]);
                o[k] = f_med3(lo, mi, hi);
            }

            __builtin_nontemporal_store(o, (v4f*)(op + (size_t)r * Ww + c0));
        }

        // Next group's rows (issued last iteration) are the 4 oldest of the
        // <=8 outstanding loads: wait <=4, then rendezvous before reuse.
        wait_async_le4();
        __syncthreads();
    }
}

extern "C" void kernel_launch(void* const* d_in, const int* in_sizes, int n_in,
                              void* d_out, int out_size, void* d_ws, size_t ws_size,
                              hipStream_t stream) {
    (void)in_sizes; (void)n_in; (void)out_size; (void)d_ws; (void)ws_size;
    const float* x = (const float*)d_in[0];
    float* out     = (float*)d_out;

    dim3 grid(PLANES, Hh / SH);   // 256 x 8 = 2048 blocks
    dim3 block(TPB);              // 128 threads = 4 wave32
    hipLaunchKernelGGL(median3x3_kernel, grid, block, 0, stream, x, out);
}